// ECVQlastdim_13322988552583
// MI455X (gfx1250) — compile-verified
//
#include <hip/hip_runtime.h>
#include <math.h>

#define N_PTS 16384
#define NCB   16
#define DIM   4
#define CBSZ  1024
#define INV_LOG2 1.4426950408889634f

typedef __attribute__((ext_vector_type(2))) float v2f;
typedef __attribute__((ext_vector_type(8))) float v8f;

// ---------------------------------------------------------------------------
// Prep kernel: one block per codebook 'a'.
//   lp[a,c] = log_softmax(logits[a,:])[c] / (-ln2)      (positive bits)
//   w [a,c] = ||cb[a,c]||^2 + lp[a,c]/lambda
// Also zeroes the 4 scalar output slots.
// ---------------------------------------------------------------------------
__global__ __launch_bounds__(256) void ecvq_prep(
    const float* __restrict__ codebook, const float* __restrict__ logits,
    const float* __restrict__ lmbda_p, float* __restrict__ ws_lp,
    float* __restrict__ ws_w, float* __restrict__ out_scalars)
{
  __shared__ float red[256];
  const int a   = blockIdx.x;
  const int tid = threadIdx.x;
  const float* lg = logits + a * CBSZ;

  float m = -__builtin_inff();
  for (int k = 0; k < 4; ++k) m = fmaxf(m, lg[tid + k * 256]);
  red[tid] = m; __syncthreads();
  for (int s = 128; s > 0; s >>= 1) {
    if (tid < s) red[tid] = fmaxf(red[tid], red[tid + s]);
    __syncthreads();
  }
  m = red[0]; __syncthreads();

  float ssum = 0.0f;
  for (int k = 0; k < 4; ++k) ssum += __expf(lg[tid + k * 256] - m);
  red[tid] = ssum; __syncthreads();
  for (int s = 128; s > 0; s >>= 1) {
    if (tid < s) red[tid] += red[tid + s];
    __syncthreads();
  }
  const float logZ   = __logf(red[0]);
  const float inv_lm = 1.0f / lmbda_p[0];

  for (int k = 0; k < 4; ++k) {
    const int c = tid + k * 256;
    const float lp_nat = lg[c] - m - logZ;
    const float bits   = -lp_nat * INV_LOG2;
    const float4 cv = *(const float4*)(codebook + (size_t)(a * CBSZ + c) * DIM);
    const float nn = cv.x * cv.x + cv.y * cv.y + cv.z * cv.z + cv.w * cv.w;
    ws_lp[a * CBSZ + c] = bits;
    ws_w [a * CBSZ + c] = nn + bits * inv_lm;
  }
  if (a == 0 && tid < 4) out_scalars[tid] = 0.0f;
}

// ---------------------------------------------------------------------------
// Main kernel: grid = (N/16/8, NCB); 256 threads = 8 waves.
// Wave w owns a 16-row x-tile; loops over 64 codeword column-tiles, each tile
// one V_WMMA_F32_16X16X4_F32:  D[m][c] = w[c] - 2*x[m,:].cb[c,:]
// B fragments and w are pre-swizzled into per-lane order in LDS so the
// unrolled-by-4 loop consumes three ds_load_b128 per 4 WMMAs.
// ---------------------------------------------------------------------------
__global__ __launch_bounds__(256) void ecvq_main(
    const float* __restrict__ x, const float* __restrict__ codebook,
    const float* __restrict__ ws_lp, const float* __restrict__ ws_w,
    float* __restrict__ x_hat, float* __restrict__ rate_out)
{
  // s_bf[l][2t+q] = -2*cb[a][t*16 + (l&15)][2*(l>>4)+q]  (per-lane B fragments)
  __shared__ __align__(16) float s_bf[32][128];   // 16 KB
  __shared__ __align__(16) float s_wt[16][64];    //  4 KB  s_wt[c%16][c/16] = w[c]
  __shared__ float s_lp[CBSZ];                    //  4 KB
  __shared__ float s_rate;

  const int a   = blockIdx.y;
  const int tid = threadIdx.x;
  if (tid == 0) s_rate = 0.0f;

  // ---- stage swizzled tables -------------------------------------------
  const float4* cb4 = (const float4*)(codebook + (size_t)a * CBSZ * DIM);
  for (int i = tid; i < CBSZ; i += 256) {
    const float4 v = cb4[i];             // codeword i: (d0,d1,d2,d3)
    const int t  = i >> 4;
    const int nl = i & 15;
    s_bf[nl     ][2 * t    ] = -2.0f * v.x;   // hi=0: K=0,1
    s_bf[nl     ][2 * t + 1] = -2.0f * v.y;
    s_bf[nl + 16][2 * t    ] = -2.0f * v.z;   // hi=1: K=2,3
    s_bf[nl + 16][2 * t + 1] = -2.0f * v.w;
    s_wt[nl][t] = ws_w [a * CBSZ + i];
    s_lp[i]     = ws_lp[a * CBSZ + i];
  }
  __syncthreads();

  const int lane  = tid & 31;
  const int nl    = lane & 15;
  const int hi    = lane >> 4;
  const int kb    = hi * 2;
  const int wave  = tid >> 5;
  const int mbase = (blockIdx.x * 8 + wave) * 16;

  // A fragment (16x4 f32): lane holds row M=nl, K = kb,kb+1 (loop-invariant).
  const float* xr = x + ((size_t)(mbase + nl) * (NCB * DIM) + a * DIM + kb);
  v2f afrag; afrag.x = xr[0]; afrag.y = xr[1];

  float bestv[8];
  int   besti[8];
#pragma unroll
  for (int j = 0; j < 8; ++j) { bestv[j] = __builtin_inff(); besti[j] = 0; }

  const float* bl = &s_bf[lane][0];
  const float* wl = &s_wt[nl][0];

  auto tile = [&](int t, float b0, float b1, float wv) {
    v2f bfrag; bfrag.x = b0; bfrag.y = b1;
    v8f cfrag;
#pragma unroll
    for (int j = 0; j < 8; ++j) cfrag[j] = wv;   // broadcast w[c] down column
    const v8f d = __builtin_amdgcn_wmma_f32_16x16x4_f32(
        false, afrag, false, bfrag, (short)0, cfrag, false, false);
    const int c = t * 16 + nl;
#pragma unroll
    for (int j = 0; j < 8; ++j) {
      if (d[j] < bestv[j]) { bestv[j] = d[j]; besti[j] = c; }
    }
  };

  for (int t4 = 0; t4 < 64; t4 += 4) {
    const float4 w4 = *(const float4*)(wl + t4);          // w for 4 tiles
    const float4 bA = *(const float4*)(bl + 2 * t4);      // frags t4, t4+1
    const float4 bB = *(const float4*)(bl + 2 * t4 + 4);  // frags t4+2, t4+3
    tile(t4 + 0, bA.x, bA.y, w4.x);
    tile(t4 + 1, bA.z, bA.w, w4.y);
    tile(t4 + 2, bB.x, bB.y, w4.z);
    tile(t4 + 3, bB.z, bB.w, w4.w);
  }

  // Reduce argmin across the 16 lanes of each half (columns c === nl mod 16).
#pragma unroll
  for (int j = 0; j < 8; ++j) {
    float v = bestv[j];
    int   i = besti[j];
#pragma unroll
    for (int msk = 1; msk <= 8; msk <<= 1) {
      const float ov = __shfl_xor(v, msk, 32);
      const int   oi = __shfl_xor(i, msk, 32);
      if (ov < v || (ov == v && oi < i)) { v = ov; i = oi; }
    }
    bestv[j] = v; besti[j] = i;
  }

  // Lanes nl<8 own row M = hi*8 + nl: gather codeword + rate contribution.
  float r = 0.0f;
  if (nl < 8) {
    const int row = hi * 8 + nl;
    const int idx = besti[nl];
    const float4 cv = cb4[idx];                   // gather winning codeword (L2)
    *(float4*)(x_hat + ((size_t)(mbase + row) * (NCB * DIM) + a * DIM)) = cv;
    r = s_lp[idx];
  }
  // rate_uem: wave reduce, block accumulate in LDS, one global atomic per block.
#pragma unroll
  for (int msk = 1; msk < 32; msk <<= 1) r += __shfl_xor(r, msk, 32);
  if (lane == 0) atomicAdd(&s_rate, r);
  __syncthreads();
  if (tid == 0) atomicAdd(rate_out, s_rate);
}

extern "C" void kernel_launch(void* const* d_in, const int* in_sizes, int n_in,
                              void* d_out, int out_size, void* d_ws, size_t ws_size,
                              hipStream_t stream) {
  (void)in_sizes; (void)n_in; (void)out_size; (void)ws_size;
  const float* x        = (const float*)d_in[0];
  const float* codebook = (const float*)d_in[1];
  const float* logits   = (const float*)d_in[2];
  const float* lmbda    = (const float*)d_in[3];

  float* out     = (float*)d_out;
  float* ws      = (float*)d_ws;
  float* ws_lp   = ws;               // NCB*CBSZ floats
  float* ws_w    = ws + NCB * CBSZ;  // NCB*CBSZ floats
  float* x_hat   = out;
  float* scalars = out + (size_t)N_PTS * NCB * DIM;  // [rate_uem, rate_cem, prior_dist, param_bit]

  ecvq_prep<<<dim3(NCB), dim3(256), 0, stream>>>(codebook, logits, lmbda,
                                                 ws_lp, ws_w, scalars);
  ecvq_main<<<dim3(N_PTS / 16 / 8, NCB), dim3(256), 0, stream>>>(
      x, codebook, ws_lp, ws_w, x_hat, scalars);
}